// UniSAGELayer_76854144795177
// MI455X (gfx1250) — compile-verified
//
#include <hip/hip_runtime.h>
#include <hip/hip_bf16.h>

typedef __attribute__((ext_vector_type(2))) float v2f;
typedef __attribute__((ext_vector_type(8))) float v8f;

#define DCH 128          // feature dim (in == out == 128)
#define LDSW 132         // padded LDS row stride (132 mod 64 = 4 -> conflict-free b64 frag loads)

// ---------------------------------------------------------------------------
// Kernel 1: x = x_0 @ W^T + b   via V_WMMA_F32_16X16X4_F32 (full fp32 path)
// Block = 128 threads = 4 waves. Each block: W in LDS once, then persistent
// loop over 16-row M tiles. Each wave computes two 16x16 output tiles.
// ---------------------------------------------------------------------------
__global__ __launch_bounds__(128) void unisage_gemm_wmma(
    const float* __restrict__ x0, const float* __restrict__ W,
    const float* __restrict__ bias, float* __restrict__ out, int n_nodes)
{
    __shared__ float sW[DCH * LDSW];   // W row-major, padded: sW[n*LDSW + k] = W[n][k]
    __shared__ float sA[16 * LDSW];    // A tile row-major, padded: sA[m*LDSW + k]

    const int tid  = threadIdx.x;
    const int lane = tid & 31;
    const int wave = tid >> 5;

    // Stage all of W (64 KB logical) into LDS once per block, coalesced.
    {
        const int n = tid;                      // 128 threads -> one W row each
        const float* src = &W[(size_t)n * DCH];
        float* dst = &sW[n * LDSW];
        #pragma unroll
        for (int k = 0; k < DCH; k += 4) {
            float4 w = *(const float4*)&src[k];
            dst[k + 0] = w.x; dst[k + 1] = w.y; dst[k + 2] = w.z; dst[k + 3] = w.w;
        }
    }
    __syncthreads();

    const int row = lane & 15;        // M index (A frag) / N index (B,C,D frags)
    const int kh  = lane >> 4;        // K-half select: lanes 16-31 hold K+2,K+3
    const int nt0 = wave * 32;        // first N tile for this wave
    const int nt1 = wave * 32 + 16;   // second N tile
    const float bias0 = bias[nt0 + row];
    const float bias1 = bias[nt1 + row];

    const int mtiles = n_nodes >> 4;  // n_nodes is a multiple of 16 (100000)

    for (int mt = blockIdx.x; mt < mtiles; mt += gridDim.x) {
        const int mbase = mt << 4;

        // Stage 16x128 A tile: thread t loads row (t>>3), 16 cols starting (t&7)*16.
        {
            const int m     = tid >> 3;
            const int cbase = (tid & 7) * 16;
            const float* src = &x0[(size_t)(mbase + m) * DCH + cbase];
            float* dst = &sA[m * LDSW + cbase];
            #pragma unroll
            for (int j = 0; j < 16; j += 4) {
                float4 v = *(const float4*)&src[j];
                dst[j + 0] = v.x; dst[j + 1] = v.y; dst[j + 2] = v.z; dst[j + 3] = v.w;
            }
        }
        __syncthreads();

        v8f acc0 = {};
        v8f acc1 = {};
        #pragma unroll
        for (int kt = 0; kt < DCH; kt += 4) {
            const int k0 = kt + kh * 2;
            // A frag: A[row][k0], A[row][k0+1]   (16x4 fp32 layout)
            v2f a = *(const v2f*)&sA[row * LDSW + k0];
            // B frag: B[k][n] = W[n][k] -> read W rows (4x16 fp32 layout)
            v2f b0 = *(const v2f*)&sW[(nt0 + row) * LDSW + k0];
            v2f b1 = *(const v2f*)&sW[(nt1 + row) * LDSW + k0];
            acc0 = __builtin_amdgcn_wmma_f32_16x16x4_f32(false, a, false, b0,
                                                         (short)0, acc0, false, false);
            acc1 = __builtin_amdgcn_wmma_f32_16x16x4_f32(false, a, false, b1,
                                                         (short)0, acc1, false, false);
        }

        // C/D layout: VGPR v -> row v (lanes 0-15) or v+8 (lanes 16-31), col = lane&15.
        const int nc0 = nt0 + row;
        const int nc1 = nt1 + row;
        #pragma unroll
        for (int v = 0; v < 8; v++) {
            const int m = mbase + v + kh * 8;
            out[(size_t)m * DCH + nc0] = acc0[v] + bias0;
            out[(size_t)m * DCH + nc1] = acc1[v] + bias1;
        }
        __syncthreads();
    }
}

// ---------------------------------------------------------------------------
// Zero-fill workspace (accumulators must be zeroed every call).
// ---------------------------------------------------------------------------
__global__ __launch_bounds__(256) void unisage_zero(float4* __restrict__ p, int n4)
{
    int i = blockIdx.x * blockDim.x + threadIdx.x;
    int stride = gridDim.x * blockDim.x;
    float4 z = {0.f, 0.f, 0.f, 0.f};
    for (; i < n4; i += stride) p[i] = z;
}

// ---------------------------------------------------------------------------
// Pass 1: m01[e] += x[v]  (one wave per nonzero; float4 per lane; L2 atomics)
// ---------------------------------------------------------------------------
__global__ __launch_bounds__(256) void unisage_scatter_e(
    const float* __restrict__ x, const int* __restrict__ node_ids,
    const int* __restrict__ edge_ids, float* __restrict__ m01, int nnz)
{
    const int w    = (blockIdx.x * blockDim.x + threadIdx.x) >> 5;
    const int lane = threadIdx.x & 31;
    if (w >= nnz) return;
    const int v = node_ids[w];
    const int e = edge_ids[w];
    const float4 val = *(const float4*)&x[(size_t)v * DCH + lane * 4];
    float* dst = &m01[(size_t)e * DCH + lane * 4];
    atomicAdd(dst + 0, val.x);
    atomicAdd(dst + 1, val.y);
    atomicAdd(dst + 2, val.z);
    atomicAdd(dst + 3, val.w);
}

// ---------------------------------------------------------------------------
// Pass 2: s[v] += m01[e],  cnt[v] += 1
// ---------------------------------------------------------------------------
__global__ __launch_bounds__(256) void unisage_scatter_v(
    const float* __restrict__ m01, const int* __restrict__ node_ids,
    const int* __restrict__ edge_ids, float* __restrict__ s,
    float* __restrict__ cnt, int nnz)
{
    const int w    = (blockIdx.x * blockDim.x + threadIdx.x) >> 5;
    const int lane = threadIdx.x & 31;
    if (w >= nnz) return;
    const int v = node_ids[w];
    const int e = edge_ids[w];
    const float4 val = *(const float4*)&m01[(size_t)e * DCH + lane * 4];
    float* dst = &s[(size_t)v * DCH + lane * 4];
    atomicAdd(dst + 0, val.x);
    atomicAdd(dst + 1, val.y);
    atomicAdd(dst + 2, val.z);
    atomicAdd(dst + 3, val.w);
    if (lane == 0) atomicAdd(&cnt[v], 1.0f);
}

// ---------------------------------------------------------------------------
// Finalize: out = x + s / max(cnt, 1)   (x already lives in d_out)
// ---------------------------------------------------------------------------
__global__ __launch_bounds__(256) void unisage_finalize(
    float* __restrict__ out, const float* __restrict__ s,
    const float* __restrict__ cnt, int n_nodes)
{
    const int i = blockIdx.x * blockDim.x + threadIdx.x;   // one float4 each
    const int total = n_nodes * (DCH / 4);
    if (i >= total) return;
    const int node = i / (DCH / 4);
    const float c = fmaxf(cnt[node], 1.0f);
    const float inv = 1.0f / c;
    float4 sv = ((const float4*)s)[i];
    float4 ov = ((float4*)out)[i];
    ov.x += sv.x * inv; ov.y += sv.y * inv; ov.z += sv.z * inv; ov.w += sv.w * inv;
    ((float4*)out)[i] = ov;
}

extern "C" void kernel_launch(void* const* d_in, const int* in_sizes, int n_in,
                              void* d_out, int out_size, void* d_ws, size_t ws_size,
                              hipStream_t stream) {
    const float* x0       = (const float*)d_in[0];
    const float* W        = (const float*)d_in[1];
    const float* bias     = (const float*)d_in[2];
    const int*   node_ids = (const int*)d_in[3];
    const int*   edge_ids = (const int*)d_in[4];

    const int n_nodes = in_sizes[0] / DCH;   // 100000
    const int nnz     = in_sizes[3];         // 800000
    const int n_edges = 50000;               // fixed by the problem definition

    float* out = (float*)d_out;
    float* m01 = (float*)d_ws;                              // [n_edges, 128]
    float* s   = m01 + (size_t)n_edges * DCH;               // [n_nodes, 128]
    float* cnt = s   + (size_t)n_nodes * DCH;               // [n_nodes]

    // 1) zero accumulators (m01 | s | cnt are contiguous)
    const int zero_floats = n_edges * DCH + n_nodes * DCH + n_nodes; // multiple of 4
    unisage_zero<<<2048, 256, 0, stream>>>((float4*)m01, zero_floats / 4);

    // 2) x = x_0 @ W^T + b  (written into d_out), fp32 WMMA
    unisage_gemm_wmma<<<512, 128, 0, stream>>>(x0, W, bias, out, n_nodes);

    // 3) hyperedge aggregation (sum)
    const int waves_blocks = (nnz + 7) / 8;   // 8 waves of 32 per 256-thread block
    unisage_scatter_e<<<waves_blocks, 256, 0, stream>>>(out, node_ids, edge_ids, m01, nnz);

    // 4) node aggregation (sum + degree count)
    unisage_scatter_v<<<waves_blocks, 256, 0, stream>>>(m01, node_ids, edge_ids, s, cnt, nnz);

    // 5) out = x + s / max(cnt,1)
    const int fin_threads = n_nodes * (DCH / 4);
    unisage_finalize<<<(fin_threads + 255) / 256, 256, 0, stream>>>(out, s, cnt, n_nodes);
}